// GravNetLayer_15822659518590
// MI455X (gfx1250) — compile-verified
//
#include <hip/hip_runtime.h>
#include <hip/hip_bf16.h>
#include <math.h>

// GravNet layer for MI455X (gfx1250), wave32 + WMMA f32 16x16x4 + async-LDS.
//
// Pipeline:
//   1) embed_kernel    : spatial = x@W_s+b_s, learned = x@W_l+b_l   (WMMA)
//   2) sqnorm_kernel   : sqn[i] = |spatial_i|^2
//   3) knn_kernel      : brute-force top-16; candidates double-buffered into
//                        LDS via global_load_async_to_lds_b128; rank-4 WMMA
//                        produces 16x16 dot tiles; register sorted top-16
//                        with a ballot-guarded (wave-uniform) skip
//   4) aggregate_kernel: w=exp(-d2), mean/max over K neighbors
//   5) output_kernel   : out = concat[x,mean,max]@W_o + b_o         (WMMA)

typedef __attribute__((ext_vector_type(2))) float v2f;
typedef __attribute__((ext_vector_type(4))) float v4f;
typedef __attribute__((ext_vector_type(8))) float v8f;
typedef __attribute__((ext_vector_type(4))) int   v4i;

#define F_IN  64
#define S_DIM 4
#define FLR   64
#define F_OUT 128
#define KNN   16

#define CH    128          // candidates staged per chunk
#define CHT   (CH / 16)    // 8 WMMA tiles per chunk
#define STAGE_OPS 5        // async instructions per chunk (4 spatial + 1 sqn)

#if __has_builtin(__builtin_amdgcn_global_load_async_to_lds_b128)
#define HAVE_ASYNC 1
// Parameter types per hipcc diagnostic: vector_size(16) int pointers,
// global (AS1) source and LDS (AS3) destination.
typedef int vi4 __attribute__((vector_size(4 * sizeof(int))));
typedef __attribute__((address_space(1))) vi4* as1_vi4p;
typedef __attribute__((address_space(3))) vi4* as3_vi4p;
#else
#define HAVE_ASYNC 0
#endif

#if __has_builtin(__builtin_amdgcn_s_wait_asynccnt)
#define WAIT_ASYNC(n) __builtin_amdgcn_s_wait_asynccnt(n)
#else
#define WAIT_ASYNC(n) asm volatile("s_wait_asynccnt %0" ::"i"(n) : "memory")
#endif

#if __has_builtin(__builtin_amdgcn_ballot_w32)
#define ANY32(p) (__builtin_amdgcn_ballot_w32(p) != 0u)
#else
#define ANY32(p) (__ballot(p) != 0ull)
#endif

static __device__ __forceinline__ v8f wmma4(v2f a, v2f b, v8f c) {
  // D = A(16x4,f32) * B(4x16,f32) + C(16x16,f32)
  return __builtin_amdgcn_wmma_f32_16x16x4_f32(
      /*neg_a=*/false, a, /*neg_b=*/false, b,
      /*c_mod=*/(short)0, c, /*reuse_a=*/false, /*reuse_b=*/false);
}

// ---------------------------------------------------------------------------
// 1) spatial / learned embedding. One wave handles a 16-row tile of x.
//    A-frag per lane: row = lane%16, K pair = {0,1} (lanes 0-15) or {2,3}.
// ---------------------------------------------------------------------------
__global__ __launch_bounds__(32)
void embed_kernel(const float* __restrict__ x,
                  const float* __restrict__ Ws, const float* __restrict__ bs,
                  const float* __restrict__ Wl, const float* __restrict__ bl,
                  float* __restrict__ spatial, float* __restrict__ learned) {
  const int lane  = threadIdx.x;
  const int r     = lane & 15;
  const int hi    = lane >> 4;
  const int row16 = blockIdx.x * 16;

  v8f accS = {};
  v8f accL[4] = {{}, {}, {}, {}};

  for (int kk = 0; kk < F_IN / 4; ++kk) {
    const int k0 = 4 * kk + 2 * hi;
    v2f a = *(const v2f*)(x + (size_t)(row16 + r) * F_IN + k0);

    // spatial: B = W_s (64x4); unconditional clamped load + select (no exec
    // divergence in the hot loop)
    const float w0 = Ws[k0 * S_DIM + (r & 3)];
    const float w1 = Ws[(k0 + 1) * S_DIM + (r & 3)];
    v2f bS;
    bS.x = (r < S_DIM) ? w0 : 0.0f;
    bS.y = (r < S_DIM) ? w1 : 0.0f;
    accS = wmma4(a, bS, accS);

    // learned: B = W_l (64x64), 4 column tiles
#pragma unroll
    for (int t = 0; t < 4; ++t) {
      const int c = 16 * t + r;
      v2f bL;
      bL.x = Wl[k0 * FLR + c];
      bL.y = Wl[(k0 + 1) * FLR + c];
      accL[t] = wmma4(a, bL, accL[t]);
    }
  }

#pragma unroll
  for (int t = 0; t < 4; ++t) {
    const int c = 16 * t + r;
    const float bb = bl[c];
#pragma unroll
    for (int v = 0; v < 8; ++v)
      learned[(size_t)(row16 + 8 * hi + v) * FLR + c] = accL[t][v] + bb;
  }
  if (r < S_DIM) {
    const float bb = bs[r];
#pragma unroll
    for (int v = 0; v < 8; ++v)
      spatial[(size_t)(row16 + 8 * hi + v) * S_DIM + r] = accS[v] + bb;
  }
}

// ---------------------------------------------------------------------------
// 2) squared norms of spatial coords
// ---------------------------------------------------------------------------
__global__ __launch_bounds__(256)
void sqnorm_kernel(const float* __restrict__ spatial, float* __restrict__ sqn,
                   int N) {
  const int i = blockIdx.x * blockDim.x + threadIdx.x;
  if (i >= N) return;
  v4f s = *(const v4f*)(spatial + (size_t)i * S_DIM);
  sqn[i] = s.x * s.x + s.y * s.y + s.z * s.z + s.w * s.w;
}

// ---------------------------------------------------------------------------
// 3) brute-force kNN. One wave per 16 queries (queries in B / columns).
//    Candidate chunks (128 rows) are double-buffered into LDS with
//    global_load_async_to_lds_b128; one f32 WMMA per 16-candidate tile.
// ---------------------------------------------------------------------------
static __device__ __forceinline__ void stage_chunk(
    const float* __restrict__ spatial, const float* __restrict__ sqn,
    int c0, float* lsp, float* lsq, int lane) {
#if HAVE_ASYNC
#pragma unroll
  for (int j = 0; j < 4; ++j) {
    const float* g = spatial + (size_t)c0 * S_DIM + (size_t)(j * 32 + lane) * 4;
    float* l = lsp + (j * 32 + lane) * 4;
    __builtin_amdgcn_global_load_async_to_lds_b128(
        (as1_vi4p)(uintptr_t)g, (as3_vi4p)(uint32_t)(uintptr_t)l, 0, 0);
  }
  {
    const float* g = sqn + c0 + lane * 4;
    float* l = lsq + lane * 4;
    __builtin_amdgcn_global_load_async_to_lds_b128(
        (as1_vi4p)(uintptr_t)g, (as3_vi4p)(uint32_t)(uintptr_t)l, 0, 0);
  }
#else
#pragma unroll
  for (int j = 0; j < 4; ++j) {
    v4f t = *(const v4f*)(spatial + (size_t)c0 * S_DIM + (size_t)(j * 32 + lane) * 4);
    *(v4f*)(lsp + (j * 32 + lane) * 4) = t;
  }
  v4f tq = *(const v4f*)(sqn + c0 + lane * 4);
  *(v4f*)(lsq + lane * 4) = tq;
#endif
}

__global__ __launch_bounds__(32)
void knn_kernel(const float* __restrict__ spatial,
                const float* __restrict__ sqn,
                int* __restrict__ knnidx, int N) {
  __shared__ __align__(16) float lsp[2][CH * S_DIM];  // staged spatial rows
  __shared__ __align__(16) float lsq[2][CH];          // staged |c|^2
  __shared__ float sd[32][KNN];
  __shared__ int   si[32][KNN];

  const int lane = threadIdx.x;
  const int r    = lane & 15;
  const int hi   = lane >> 4;
  const int q0   = blockIdx.x * 16;
  const int qcol = q0 + r;

  // B fragment: B[k][n] = spatial[query n][k]
  v2f b = *(const v2f*)(spatial + (size_t)qcol * S_DIM + 2 * hi);
  const float qs = sqn[qcol];

  float bestd[KNN];
  int   besti[KNN];
#pragma unroll
  for (int t = 0; t < KNN; ++t) { bestd[t] = __builtin_inff(); besti[t] = 0; }

  const int ntiles = N >> 4;                       // 3125
  const int nch    = (ntiles + CHT - 1) / CHT;     // 391 (padded reads stay in ws)

  // explicit double-buffer pointers (avoid loop specialization on ch&1)
  float* cur_sp = lsp[0]; float* cur_sq = lsq[0];
  float* nxt_sp = lsp[1]; float* nxt_sq = lsq[1];

  stage_chunk(spatial, sqn, 0, cur_sp, cur_sq, lane);

#pragma clang loop unroll(disable)
  for (int ch = 0; ch < nch; ++ch) {
    if (ch + 1 < nch) {
      stage_chunk(spatial, sqn, (ch + 1) * CH, nxt_sp, nxt_sq, lane);
      WAIT_ASYNC(STAGE_OPS);  // chunk `ch` landed; next chunk still in flight
    } else {
      WAIT_ASYNC(0);
    }
    asm volatile("" ::: "memory");

    const int c0 = ch * CH;
#pragma clang loop unroll(disable)
    for (int t = 0; t < CHT; ++t) {
      // A fragment from LDS: A[m][k] = staged candidate spatial
      v2f a = *(const v2f*)&cur_sp[(t * 16 + r) * S_DIM + 2 * hi];
      const int mbase = c0 + t * 16 + 8 * hi;
      v4f cs0 = *(const v4f*)&cur_sq[t * 16 + 8 * hi];
      v4f cs1 = *(const v4f*)&cur_sq[t * 16 + 8 * hi + 4];

      v8f acc = {};
      acc = wmma4(a, b, acc);  // acc[v] = cand(mbase+v) . query(qcol)

      const float cs[8] = {cs0.x, cs0.y, cs0.z, cs0.w,
                           cs1.x, cs1.y, cs1.z, cs1.w};
      float d2v[8];
#pragma unroll
      for (int v = 0; v < 8; ++v) {
        const int m = mbase + v;
        float d2 = cs[v] + qs - 2.0f * acc[v];
        d2v[v] = (m == qcol || m >= N) ? __builtin_inff() : d2;  // self / pad
      }
      // wave-uniform skip: most tiles cannot improve anyone's top-16
      float mn8 = d2v[0];
#pragma unroll
      for (int v = 1; v < 8; ++v) mn8 = fminf(mn8, d2v[v]);
      if (ANY32(mn8 < bestd[KNN - 1])) {
#pragma unroll
        for (int v = 0; v < 8; ++v) {
          if (d2v[v] < bestd[KNN - 1]) {
            float dd = d2v[v]; int ii = mbase + v;
#pragma unroll
            for (int s = 0; s < KNN; ++s) {
              if (dd < bestd[s]) {
                const float td = bestd[s]; const int ti = besti[s];
                bestd[s] = dd; besti[s] = ii;
                dd = td; ii = ti;
              }
            }
          }
        }
      }
    }

    // swap double buffers
    float* tsp = cur_sp; cur_sp = nxt_sp; nxt_sp = tsp;
    float* tsq = cur_sq; cur_sq = nxt_sq; nxt_sq = tsq;
  }

#pragma unroll
  for (int t = 0; t < KNN; ++t) { sd[lane][t] = bestd[t]; si[lane][t] = besti[t]; }
  __syncthreads();

  if (lane < 16) {
    int pa = 0, pb = 0;
#pragma unroll
    for (int t = 0; t < KNN; ++t) {
      const float da = sd[lane][pa];
      const float db = sd[lane + 16][pb];
      const bool sel = da <= db;
      knnidx[(size_t)(q0 + lane) * KNN + t] = sel ? si[lane][pa] : si[lane + 16][pb];
      pa += sel ? 1 : 0;
      pb += sel ? 0 : 1;
    }
  }
}

// ---------------------------------------------------------------------------
// 4) gaussian-weighted mean/max aggregation over the 16 neighbors
// ---------------------------------------------------------------------------
__global__ __launch_bounds__(256)
void aggregate_kernel(const float* __restrict__ spatial,
                      const float* __restrict__ learned,
                      const int* __restrict__ knnidx,
                      float* __restrict__ meanb, float* __restrict__ maxb,
                      int N) {
  const int gid = blockIdx.x * blockDim.x + threadIdx.x;
  if (gid >= N * FLR) return;
  const int i = gid >> 6;   // node
  const int f = gid & 63;   // feature

  v4i idx[4];
#pragma unroll
  for (int j = 0; j < 4; ++j)
    idx[j] = *(const v4i*)(knnidx + (size_t)i * KNN + 4 * j);
  int nb[KNN];
#pragma unroll
  for (int j = 0; j < 4; ++j) {
    nb[4 * j + 0] = idx[j].x; nb[4 * j + 1] = idx[j].y;
    nb[4 * j + 2] = idx[j].z; nb[4 * j + 3] = idx[j].w;
  }

  v4f qi = *(const v4f*)(spatial + (size_t)i * S_DIM);
  float sum  = 0.0f;
  float mval = -__builtin_inff();
#pragma unroll
  for (int k = 0; k < KNN; ++k) {
    v4f ns = *(const v4f*)(spatial + (size_t)nb[k] * S_DIM);
    const float dx = qi.x - ns.x, dy = qi.y - ns.y;
    const float dz = qi.z - ns.z, dw = qi.w - ns.w;
    const float d2 = dx * dx + dy * dy + dz * dz + dw * dw;
    const float w  = __expf(-d2);
    const float v  = learned[(size_t)nb[k] * FLR + f] * w;
    sum += v;
    mval = fmaxf(mval, v);
  }
  meanb[gid] = sum * (1.0f / KNN);
  maxb[gid]  = mval;
}

// ---------------------------------------------------------------------------
// 5) out = concat[x, mean, max] @ W_o + b_o  (192 -> 128), WMMA 16x16x4.
//    Bias folded into the C-init (C columns are per-lane).
// ---------------------------------------------------------------------------
__global__ __launch_bounds__(32)
void output_kernel(const float* __restrict__ x,
                   const float* __restrict__ meanb,
                   const float* __restrict__ maxb,
                   const float* __restrict__ Wo,
                   const float* __restrict__ bo,
                   float* __restrict__ out) {
  const int lane  = threadIdx.x;
  const int r     = lane & 15;
  const int hi    = lane >> 4;
  const int row16 = blockIdx.x * 16;
  const int rowg  = row16 + r;

  const int KTOT = (F_IN + 2 * FLR) / 4;  // 48 K-steps of 4
  v2f afrag[48];
#pragma unroll
  for (int kk = 0; kk < 48; ++kk) {
    const int k0 = 4 * kk + 2 * hi;
    const float* p;
    int kl = k0;
    if (k0 < 64)       { p = x     + (size_t)rowg * 64;                }
    else if (k0 < 128) { p = meanb + (size_t)rowg * 64; kl = k0 - 64;  }
    else               { p = maxb  + (size_t)rowg * 64; kl = k0 - 128; }
    afrag[kk] = *(const v2f*)(p + kl);
  }

  for (int t = 0; t < 8; ++t) {
    const int c = 16 * t + r;
    const float bb = bo[c];
    v8f acc = {bb, bb, bb, bb, bb, bb, bb, bb};
#pragma unroll
    for (int kk = 0; kk < KTOT; ++kk) {
      const int k0 = 4 * kk + 2 * hi;
      v2f b;
      b.x = Wo[k0 * F_OUT + c];
      b.y = Wo[(k0 + 1) * F_OUT + c];
      acc = wmma4(afrag[kk], b, acc);
    }
#pragma unroll
    for (int v = 0; v < 8; ++v)
      out[(size_t)(row16 + 8 * hi + v) * F_OUT + c] = acc[v];
  }
}

// ---------------------------------------------------------------------------
extern "C" void kernel_launch(void* const* d_in, const int* in_sizes, int n_in,
                              void* d_out, int out_size, void* d_ws, size_t ws_size,
                              hipStream_t stream) {
  const float* x  = (const float*)d_in[0];
  const float* Ws = (const float*)d_in[1];
  const float* bs = (const float*)d_in[2];
  const float* Wl = (const float*)d_in[3];
  const float* bl = (const float*)d_in[4];
  const float* Wo = (const float*)d_in[5];
  const float* bo = (const float*)d_in[6];
  float* out = (float*)d_out;

  const int N = in_sizes[0] / F_IN;   // 50000

  // workspace carving (floats), ~43 MB total
  float* spatial = (float*)d_ws;                          // N*4
  float* sqn     = spatial + (size_t)N * S_DIM;           // N
  float* learned = sqn + N;                               // N*64
  int*   knnbuf  = (int*)(learned + (size_t)N * FLR);     // N*16
  float* meanb   = (float*)(knnbuf + (size_t)N * KNN);    // N*64
  float* maxb    = meanb + (size_t)N * FLR;               // N*64

  const int tiles = N / 16;  // 3125

  embed_kernel<<<tiles, 32, 0, stream>>>(x, Ws, bs, Wl, bl, spatial, learned);
  sqnorm_kernel<<<(N + 255) / 256, 256, 0, stream>>>(spatial, sqn, N);
  knn_kernel<<<tiles, 32, 0, stream>>>(spatial, sqn, knnbuf, N);
  aggregate_kernel<<<(N * FLR + 255) / 256, 256, 0, stream>>>(
      spatial, learned, knnbuf, meanb, maxb, N);
  output_kernel<<<tiles, 32, 0, stream>>>(x, meanb, maxb, Wo, bo, out);
}